// WeightAndSum_79388175499518
// MI455X (gfx1250) — compile-verified
//
#include <hip/hip_runtime.h>
#include <hip/hip_bf16.h>

// ---------------------------------------------------------------------------
// WeightAndSum pooling for MI455X (gfx1250, wave32)
//   gate = sigmoid(feats @ W + b)   -> v_wmma_f32_16x16x4_f32 chain
//   out[seg] += gate * feats        -> run-length accumulation (sorted idx)
// Data path: global -> LDS via global_load_async_to_lds_b128 (double
// buffered, ASYNCcnt-throttled); LDS feeds both the WMMA A-matrix layout and
// the per-lane accumulation reads. HBM roofline: 1.02 GB read @ 23.3 TB/s.
// ---------------------------------------------------------------------------

typedef __attribute__((ext_vector_type(2))) float v2f;
typedef __attribute__((ext_vector_type(8))) float v8f;
typedef int b128_t __attribute__((vector_size(16)));   // matches builtin sig

#define N_FEATS          256
#define NODES_PER_GROUP  16                       // one WMMA M-tile
#define GROUPS_PER_WAVE  4
#define NODES_PER_WAVE   (NODES_PER_GROUP * GROUPS_PER_WAVE)   // 64
#define WAVES_PER_BLOCK  4
#define BLOCK_THREADS    (WAVES_PER_BLOCK * 32)                // 128
#define NODES_PER_BLOCK  (NODES_PER_WAVE * WAVES_PER_BLOCK)    // 256
#define TILE_FLOATS      (NODES_PER_GROUP * N_FEATS)           // 4096 (16 KB)

#if defined(__has_builtin)
#if __has_builtin(__builtin_amdgcn_global_load_async_to_lds_b128) && \
    __has_builtin(__builtin_amdgcn_s_wait_asynccnt)
#define USE_ASYNC_LDS 1
#endif
#endif

#define AS_GLOBAL __attribute__((address_space(1)))
#define AS_LDS    __attribute__((address_space(3)))

// Temporal hint NT (gfx12 CPol TH[2:0]=1): feats is streamed exactly once;
// keep L2 capacity for the atomically-accumulated 51 MB output instead.
#define ASYNC_CPOL 1

// Stage one 16-node x 256-feat tile (16 KB) into LDS. 32 x b128 per lane.
// Out-of-range nodes are clamped to a valid row (their results are discarded).
__device__ __forceinline__ void stage_group(const float* __restrict__ feats,
                                            int nodeBase, int nNodes,
                                            float* tile, int lane) {
#pragma unroll
  for (int t = 0; t < 32; ++t) {
    const int f4 = t * 32 + lane;            // float4 index in tile [0,1024)
    int node = nodeBase + (f4 >> 6);         // 64 float4 per node row
    if (node >= nNodes) node = nNodes - 1;   // clamp: safe dummy read
    const float* src = feats + (size_t)node * N_FEATS + (f4 & 63) * 4;
    float* dst = tile + f4 * 4;
#if defined(USE_ASYNC_LDS)
    __builtin_amdgcn_global_load_async_to_lds_b128(
        (AS_GLOBAL b128_t*)src, (AS_LDS b128_t*)dst, 0, ASYNC_CPOL);
#else
    *(float4*)dst = *(const float4*)src;
#endif
  }
}

__device__ __forceinline__ void flush_seg(float* __restrict__ out, int seg,
                                          int lane, const float4& a0,
                                          const float4& a1) {
  float* dst = out + (size_t)seg * N_FEATS + lane * 8;
  unsafeAtomicAdd(dst + 0, a0.x);
  unsafeAtomicAdd(dst + 1, a0.y);
  unsafeAtomicAdd(dst + 2, a0.z);
  unsafeAtomicAdd(dst + 3, a0.w);
  unsafeAtomicAdd(dst + 4, a1.x);
  unsafeAtomicAdd(dst + 5, a1.y);
  unsafeAtomicAdd(dst + 6, a1.z);
  unsafeAtomicAdd(dst + 7, a1.w);
}

// Gate GEMV via WMMA + run-length weighted accumulation for one 16-node tile.
__device__ __forceinline__ void process_group(
    const float* tile, const float* wlds, float bias,
    const int* __restrict__ idx, int nodeBase, int nNodes, int lane,
    float* __restrict__ out, int& curSeg, float4& acc0, float4& acc1) {
  // ---- gate logits: D[16x16] = A[16x4-chunks] x Wbcast + bias ----
  v8f c;
#pragma unroll
  for (int i = 0; i < 8; ++i) c[i] = bias;

  const int halfSel = (lane >> 4) << 1;          // 0 for lanes 0-15, 2 for 16-31
  const int arow = (lane & 15) * N_FEATS;        // A: M = lane%16
#pragma unroll 4
  for (int kb = 0; kb < N_FEATS; kb += 4) {
    v2f a = *(const v2f*)(tile + arow + kb + halfSel);   // A[m][kb..kb+1 / +2..3]
    v2f b = *(const v2f*)(wlds + kb + halfSel);          // B col-broadcast of W
    c = __builtin_amdgcn_wmma_f32_16x16x4_f32(false, a, false, b,
                                              (short)0, c, false, false);
  }

  // ---- weighted accumulation, batched by segment run ----
#pragma unroll
  for (int m = 0; m < NODES_PER_GROUP; ++m) {
    const int node = nodeBase + m;
    if (node >= nNodes) break;                       // wave-uniform
    // D layout: VGPR r holds M=r (lanes 0-15) and M=r+8 (lanes 16-31).
    // Source lane is a compile-time constant -> v_readlane_b32 broadcast.
    const float logit = __uint_as_float((unsigned)__builtin_amdgcn_readlane(
        (int)__float_as_uint(c[m & 7]), (m >> 3) * 16));
    // fast sigmoid: v_exp_f32 + v_rcp_f32 (no IEEE divide sequence)
    const float gate = __builtin_amdgcn_rcpf(1.0f + __expf(-logit));
    const int seg = idx[node];                       // sorted, wave-uniform
    if (seg != curSeg) {
      if (curSeg >= 0) flush_seg(out, curSeg, lane, acc0, acc1);
      acc0 = make_float4(0.f, 0.f, 0.f, 0.f);
      acc1 = make_float4(0.f, 0.f, 0.f, 0.f);
      curSeg = seg;
    }
    const float* f = tile + m * N_FEATS + lane * 8;
    const float4 f0 = *(const float4*)(f);
    const float4 f1 = *(const float4*)(f + 4);
    acc0.x += gate * f0.x; acc0.y += gate * f0.y;
    acc0.z += gate * f0.z; acc0.w += gate * f0.w;
    acc1.x += gate * f1.x; acc1.y += gate * f1.y;
    acc1.z += gate * f1.z; acc1.w += gate * f1.w;
  }
}

extern "C" __global__ void __launch_bounds__(BLOCK_THREADS)
weight_and_sum_kernel(const float* __restrict__ feats,
                      const int* __restrict__ idx,
                      const float* __restrict__ W,
                      const float* __restrict__ bias,
                      float* __restrict__ out, int nNodes) {
  extern __shared__ float smem[];   // [waves*2*TILE_FLOATS] tiles | [256] W
  const int tid = threadIdx.x;
  const int lane = tid & 31;
  const int wave = tid >> 5;

  float* wlds = smem + WAVES_PER_BLOCK * 2 * TILE_FLOATS;
  if (tid < 128) {                  // stage W (256 floats) once per block
    wlds[tid] = W[tid];
    wlds[tid + 128] = W[tid + 128];
  }
  __syncthreads();

  const float b = bias[0];
  const int stripBase = (blockIdx.x * WAVES_PER_BLOCK + wave) * NODES_PER_WAVE;
  if (stripBase >= nNodes) return;

  float* buf0 = smem + wave * 2 * TILE_FLOATS;
  float* buf1 = buf0 + TILE_FLOATS;

  stage_group(feats, stripBase, nNodes, buf0, lane);   // prologue fill

  int curSeg = -1;
  float4 acc0 = make_float4(0.f, 0.f, 0.f, 0.f);
  float4 acc1 = make_float4(0.f, 0.f, 0.f, 0.f);

  for (int g = 0; g < GROUPS_PER_WAVE; ++g) {
    const int gBase = stripBase + g * NODES_PER_GROUP;
    if (gBase >= nNodes) break;
    float* cur = (g & 1) ? buf1 : buf0;
    float* nxt = (g & 1) ? buf0 : buf1;
    const int nBase = gBase + NODES_PER_GROUP;
    const bool haveNext = (g + 1 < GROUPS_PER_WAVE) && (nBase < nNodes);
    if (haveNext) stage_group(feats, nBase, nNodes, nxt, lane);  // prefetch
#if defined(USE_ASYNC_LDS)
    // async-to-LDS completes in order: <=32 outstanding => current tile done
    if (haveNext) __builtin_amdgcn_s_wait_asynccnt(32);
    else          __builtin_amdgcn_s_wait_asynccnt(0);
#endif
    process_group(cur, wlds, b, idx, gBase, nNodes, lane, out, curSeg, acc0,
                  acc1);
  }
  if (curSeg >= 0) flush_seg(out, curSeg, lane, acc0, acc1);
}

extern "C" __global__ void zero_out_kernel(float4* out, int n4) {
  const int i = blockIdx.x * blockDim.x + threadIdx.x;
  if (i < n4) out[i] = make_float4(0.f, 0.f, 0.f, 0.f);
}

extern "C" void kernel_launch(void* const* d_in, const int* in_sizes, int n_in,
                              void* d_out, int out_size, void* d_ws,
                              size_t ws_size, hipStream_t stream) {
  (void)n_in; (void)d_ws; (void)ws_size;
  const float* feats = (const float*)d_in[0];
  const int* idx     = (const int*)d_in[1];   // harness passes integer as int32
  const float* W     = (const float*)d_in[2];
  const float* bias  = (const float*)d_in[3];
  float* out = (float*)d_out;
  const int nNodes = in_sizes[1];             // 1,000,000

  // zero output (atomic accumulation target; empty segments must be 0)
  const int n4 = out_size / 4;
  zero_out_kernel<<<(n4 + 255) / 256, 256, 0, stream>>>((float4*)out, n4);

  const int blocks = (nNodes + NODES_PER_BLOCK - 1) / NODES_PER_BLOCK;
  const size_t shmem =
      (size_t)(WAVES_PER_BLOCK * 2 * TILE_FLOATS + N_FEATS) * sizeof(float);
  weight_and_sum_kernel<<<blocks, BLOCK_THREADS, shmem, stream>>>(
      feats, idx, W, bias, out, nNodes);
}